// ODEINDLayer_37675453120663
// MI455X (gfx1250) — compile-verified
//
#include <hip/hip_runtime.h>

// Problem constants (from reference)
#define BSZ     16
#define N_IND   64
#define NPROB   (BSZ * N_IND)        // 1024 independent problems
#define T       50                   // N_STEP
#define O1      3                    // ORDER + 1
#define NVAR    151                  // 1 + T*O1
#define M_EQ    50
#define M_IV    2
#define M_ROWS  248                  // 50 + 2 + 49*4
#define NPAD    256                  // padded KKT size (16 blocks of 16)
#define PITCH   257                  // LDS row pitch (bank-conflict avoidance)
#define WP      17                   // Winv row pitch
#define NNZ     8                    // max nonzeros per A row (real max is 5)
#define GAMMA_C 0.5f
#define ALPHA_C 0.1f

typedef __attribute__((ext_vector_type(2))) float v2f;
typedef __attribute__((ext_vector_type(8))) float v8f;

extern "C" __global__ __launch_bounds__(256)
void ode_kkt_kernel(const float* __restrict__ coeffs,  // (B,T,O1)
                    const float* __restrict__ rhs,     // (B,T)
                    const float* __restrict__ iv_rhs,  // (B,2)
                    const float* __restrict__ steps,   // (B,T-1)
                    float* __restrict__ out)
{
    // Dynamic LDS: 256x256 KKT matrix with pitch 257 (257 KB)
    extern __shared__ float S[];
    __shared__ int   cidx[M_ROWS][NNZ];
    __shared__ float cval[M_ROWS][NNZ];
    __shared__ float Winv[16][16 * WP];  // per-block inverse of Lkk (17.4 KB)
    __shared__ float sv[NPAD];           // rhs / running residual
    __shared__ float yv[NPAD];           // forward-solve result
    __shared__ float lamv[NPAD];         // lambda
    __shared__ float xv[NVAR + 1];       // x = A^T lambda
    __shared__ float hstep[T - 1];

    const int p    = blockIdx.x;     // problem id
    const int tid  = threadIdx.x;    // 0..255 (8 wave32s)
    const int lane = tid & 31;
    const int wid  = tid >> 5;

    // ---------------- load per-problem vectors -------------------------
    if (tid < T - 1) hstep[tid] = steps[p * (T - 1) + tid];
    sv[tid] = (tid < M_EQ)        ? rhs[p * T + tid]
            : (tid < M_EQ + M_IV) ? iv_rhs[p * M_IV + (tid - M_EQ)]
            : 0.0f;
    for (int i = tid; i < NPAD * PITCH; i += 256) S[i] = 0.0f;
    __syncthreads();

    // ---------------- build sparse rows of A (<=5 nnz each) ------------
    if (tid < M_ROWS) {
        int   c[NNZ]; float v[NNZ];
        #pragma unroll
        for (int e = 0; e < NNZ; ++e) { c[e] = -1; v[e] = 0.0f; }
        if (tid < M_EQ) {
            const float* cf = coeffs + (p * T + tid) * O1;
            #pragma unroll
            for (int m = 0; m < O1; ++m) { c[m] = 1 + tid * O1 + m; v[m] = cf[m]; }
        } else if (tid < M_EQ + M_IV) {
            c[0] = 1 + (tid - M_EQ); v[0] = 1.0f;
        } else {
            const int d = tid - (M_EQ + M_IV);
            const int t = d >> 2, q = d & 3, j = q >> 1, par = q & 1;
            const float h = hstep[t];
            const int b0 = 1 + O1 * t, b1 = 1 + O1 * (t + 1);
            int ne = 0;
            c[ne] = 0; v[ne] = GAMMA_C; ++ne;               // eps column
            if (par == 0) {
                c[ne] = b1 + j; v[ne] = -GAMMA_C; ++ne;
                float hp = 1.0f;
                for (int k = j; k <= 2; ++k) {              // h^(k-j)/(k-j)!
                    c[ne] = b0 + k; v[ne] = GAMMA_C * hp; ++ne;
                    hp *= h / (float)(k - j + 1);
                }
            } else {
                c[ne] = b0 + j; v[ne] = -GAMMA_C; ++ne;
                float hp = 1.0f;
                for (int k = j; k <= 2; ++k) {              // (-h)^(k-j)/(k-j)!
                    c[ne] = b1 + k; v[ne] = GAMMA_C * hp; ++ne;
                    hp *= -h / (float)(k - j + 1);
                }
            }
        }
        #pragma unroll
        for (int e = 0; e < NNZ; ++e) { cidx[tid][e] = c[e]; cval[tid][e] = v[e]; }
    }
    __syncthreads();

    // ---------------- AAT = A*A^T + alpha*I (sparse-sparse dots) -------
    for (int e = tid; e < M_ROWS * M_ROWS; e += 256) {
        const int i = e / M_ROWS, j = e - i * M_ROWS;
        float acc = (i == j) ? ALPHA_C : 0.0f;
        #pragma unroll
        for (int a = 0; a < NNZ; ++a) {
            const int   ca = cidx[i][a];
            const float va = cval[i][a];
            #pragma unroll
            for (int b = 0; b < NNZ; ++b)
                if (cidx[j][b] == ca) acc += va * cval[j][b];   // pads add 0
        }
        S[i * PITCH + j] = acc;
    }
    if (tid < NPAD - M_ROWS)                                    // identity padding
        S[(M_ROWS + tid) * PITCH + (M_ROWS + tid)] = 1.0f;
    __syncthreads();

    // ---------------- blocked Cholesky, 16x16 blocks -------------------
    for (int kb = 0; kb < 16; ++kb) {
        const int c0 = kb * 16;

        // (1) wave 0 factors the diagonal block in place (LDS ops from a
        //     single wave complete in order -> no workgroup barriers) and
        //     computes Winv[kb] = Lkk^-1 (one column per lane, unrolled).
        if (wid == 0) {
            for (int cc = 0; cc < 16; ++cc) {
                const float dg = sqrtf(S[(c0 + cc) * PITCH + c0 + cc]);
                if (lane == cc) S[(c0 + cc) * PITCH + c0 + cc] = dg;
                if (lane > cc && lane < 16)
                    S[(c0 + lane) * PITCH + c0 + cc] =
                        S[(c0 + lane) * PITCH + c0 + cc] / dg;
                __builtin_amdgcn_wave_barrier();
                for (int j = cc + 1; j < 16; ++j)
                    if (lane >= j && lane < 16)
                        S[(c0 + lane) * PITCH + c0 + j] -=
                            S[(c0 + lane) * PITCH + c0 + cc] *
                            S[(c0 + j) * PITCH + c0 + cc];
                __builtin_amdgcn_wave_barrier();
            }
            if (lane < 16) {                 // forward-substitute column 'lane'
                const int j = lane;
                float w[16];
                #pragma unroll
                for (int r = 0; r < 16; ++r) {
                    float acc = (r == j) ? 1.0f : 0.0f;
                    #pragma unroll
                    for (int m = 0; m < 16; ++m)
                        if (m < r) acc -= S[(c0 + r) * PITCH + c0 + m] * w[m];
                    w[r] = (r < j) ? 0.0f : acc / S[(c0 + r) * PITCH + c0 + r];
                    Winv[kb][r * WP + j] = w[r];
                }
            }
        }
        __syncthreads();

        // (2) panel solve as GEMM: X = A_panel * Winv^T   (WMMA f32)
        {
            const int n   = lane & 15;
            const int mhi = (lane >> 4) << 3;
            const int kh  = (lane >> 4) << 1;
            for (int ib = kb + 1 + wid; ib <= 15; ib += 8) {   // wave-uniform
                const int Ri = ib * 16;
                v8f xfrag = {};
                #pragma unroll
                for (int kc = 0; kc < 4; ++kc) {
                    const int kl = kc * 4 + kh;
                    v2f afrag, bfrag;
                    afrag.x = S[(Ri + n) * PITCH + c0 + kl];
                    afrag.y = S[(Ri + n) * PITCH + c0 + kl + 1];
                    bfrag.x = Winv[kb][n * WP + kl];           // B = Winv^T
                    bfrag.y = Winv[kb][n * WP + kl + 1];
                    xfrag = __builtin_amdgcn_wmma_f32_16x16x4_f32(
                        false, afrag, false, bfrag, (short)0, xfrag, false, false);
                }
                #pragma unroll
                for (int v = 0; v < 8; ++v)
                    S[(Ri + mhi + v) * PITCH + c0 + n] = xfrag[v];
            }
        }
        __syncthreads();

        // (3) trailing update A[ib][jb] -= L[ib][kb]*L[jb][kb]^T  (WMMA f32)
        {
            const int nb     = 15 - kb;
            const int npairs = nb * (nb + 1) / 2;     // wave-uniform
            const int n   = lane & 15;
            const int mhi = (lane >> 4) << 3;
            const int kh  = (lane >> 4) << 1;
            for (int pr = wid; pr < npairs; pr += 8) {
                int ib = kb + 1, base = 0;
                while (pr >= base + (ib - kb)) { base += (ib - kb); ++ib; }
                const int jb = kb + 1 + (pr - base);
                const int Ri = ib * 16, Cj = jb * 16;

                v8f pfrag = {};                       // P = Lik * Ljk^T
                #pragma unroll
                for (int kc = 0; kc < 4; ++kc) {
                    const int kl = kc * 4 + kh;
                    v2f afrag, bfrag;
                    afrag.x = S[(Ri + n) * PITCH + c0 + kl];
                    afrag.y = S[(Ri + n) * PITCH + c0 + kl + 1];
                    bfrag.x = S[(Cj + n) * PITCH + c0 + kl];   // B = Ljk^T
                    bfrag.y = S[(Cj + n) * PITCH + c0 + kl + 1];
                    pfrag = __builtin_amdgcn_wmma_f32_16x16x4_f32(
                        false, afrag, false, bfrag, (short)0, pfrag, false, false);
                }
                #pragma unroll
                for (int v = 0; v < 8; ++v) {         // C <- C - P (outside chain)
                    const int idx = (Ri + mhi + v) * PITCH + Cj + n;
                    S[idx] = S[idx] - pfrag[v];
                }
            }
        }
        __syncthreads();
    }

    // ---------------- blocked forward solve L y = b --------------------
    for (int kb = 0; kb < 16; ++kb) {
        const int c0 = kb * 16;
        if (tid < 16) {                       // y_k = Winv[kb] * s_k
            float acc = 0.0f;
            #pragma unroll
            for (int c = 0; c < 16; ++c)
                acc += Winv[kb][tid * WP + c] * sv[c0 + c];
            yv[c0 + tid] = acc;
        }
        __syncthreads();
        const int r = c0 + 16 + tid;          // rank-16 residual update
        if (r < NPAD) {
            float acc = sv[r];
            #pragma unroll
            for (int c = 0; c < 16; ++c)
                acc -= S[r * PITCH + c0 + c] * yv[c0 + c];
            sv[r] = acc;
        }
        __syncthreads();
    }
    sv[tid] = yv[tid];
    __syncthreads();

    // ---------------- blocked backward solve L^T lam = y ---------------
    for (int kb = 15; kb >= 0; --kb) {
        const int c0 = kb * 16;
        if (tid < 16) {                       // lam_k = Winv[kb]^T * s_k
            float acc = 0.0f;
            #pragma unroll
            for (int c = 0; c < 16; ++c)
                acc += Winv[kb][c * WP + tid] * sv[c0 + c];
            lamv[c0 + tid] = acc;
        }
        __syncthreads();
        if (tid < c0) {                       // rows above: s -= L^T lam_k
            float acc = sv[tid];
            #pragma unroll
            for (int c = 0; c < 16; ++c)
                acc -= S[(c0 + c) * PITCH + tid] * lamv[c0 + c];
            sv[tid] = acc;
        }
        __syncthreads();
    }

    // ---------------- x = A^T lam (sparse scatter via LDS atomics) -----
    if (tid < NVAR) xv[tid] = 0.0f;
    __syncthreads();
    if (tid < M_ROWS) {
        const float l = lamv[tid];
        #pragma unroll
        for (int e = 0; e < NNZ; ++e) {
            const int c = cidx[tid][e];
            if (c >= 0) atomicAdd(&xv[c], cval[tid][e] * l);   // ds_add_f32
        }
    }
    __syncthreads();

    // ---------------- outputs ------------------------------------------
    float* u0  = out;
    float* u1  = out + (size_t)NPROB * T;
    float* u2  = out + (size_t)2 * NPROB * T;
    float* eps = out + (size_t)3 * NPROB * T;
    float* ost = out + (size_t)3 * NPROB * T + NPROB;
    if (tid < T) {
        u0[p * T + tid] = xv[1 + O1 * tid + 0];
        u1[p * T + tid] = xv[1 + O1 * tid + 1];
        u2[p * T + tid] = xv[1 + O1 * tid + 2];
    }
    if (tid == 0) eps[p] = xv[0];
    if (tid < T - 1) ost[p * (T - 1) + tid] = hstep[tid];
}

extern "C" void kernel_launch(void* const* d_in, const int* in_sizes, int n_in,
                              void* d_out, int out_size, void* d_ws, size_t ws_size,
                              hipStream_t stream) {
    (void)in_sizes; (void)n_in; (void)out_size; (void)d_ws; (void)ws_size;
    const float* coeffs = (const float*)d_in[0];
    const float* rhs    = (const float*)d_in[1];
    const float* ivr    = (const float*)d_in[2];
    const float* steps  = (const float*)d_in[3];
    float* out = (float*)d_out;
    const size_t shmem = (size_t)NPAD * PITCH * sizeof(float);  // 257 KB dynamic LDS
    ode_kkt_kernel<<<NPROB, 256, shmem, stream>>>(coeffs, rhs, ivr, steps, out);
}